// mk_79413945303851
// MI455X (gfx1250) — compile-verified
//
#include <hip/hip_runtime.h>

#define NLOC 10
#define NGI  13

typedef float v2f __attribute__((ext_vector_type(2)));
typedef float v8f __attribute__((ext_vector_type(8)));

// xor-lane exchange within wave32 via LDS swizzle hardware (no LDS memory
// touched, no address setup): group-of-32 mode, data[lane ^ MASK].
template <int MASK>
__device__ __forceinline__ float swz_xor(float x) {
    return __int_as_float(__builtin_amdgcn_ds_swizzle(
        __float_as_int(x), (MASK << 10) | 0x1f));
}

// One wave32 per element. Computes
//   nxnx  = nn/dt + k*(nx0*W*nx0^T + nx1*W*nx1^T)   (10x10, padded to 16x16)
//   bdiagA = nxnx ; diagA = diag(nxnx)
//   r0_out = r0_in + (nn/dt)*c_n - nxnx*c_i
// with V_WMMA_F32_16X16X4_F32 (full fp32 accumulation, matches reference).
//
// Padding (branch-free): rows/cols 10..15 clamp the load row to 9 (in-bounds,
// only feeds unused D rows/cols); K=13..15 only appear in the last k-step,
// where K=12 (half 0) is the sole live value, handled by one 0/1 multiply.
//
// Cache policy: stream-once data (nx, detwei, c_*, r0, all outputs) uses
// non-temporal hints; nmat (shared by every wave) stays cached.
//
// Output path: the 16x16 D tile is staged through wave-private LDS
// (unconditional ds_stores, same-wave in-order, no barrier), then the 100
// valid floats go out as contiguous, coalesced NT stores.
__global__ __launch_bounds__(256) void fem_assemble(
    const float* __restrict__ r0_in,    // [nele*NLOC]
    const float* __restrict__ c_i,      // [nele*NLOC]
    const float* __restrict__ c_n,      // [nele*NLOC]
    const float* __restrict__ kptr,     // [1]
    const float* __restrict__ dtptr,    // [1]
    const float* __restrict__ nmat,     // [NLOC][NGI] shared shape functions
    const float* __restrict__ nx,       // [nele][2][NLOC][NGI]
    const float* __restrict__ detwei,   // [nele][NGI]
    float* __restrict__ bdiagA,         // [nele][NLOC][NLOC]
    float* __restrict__ diagA,          // [nele][NLOC]
    float* __restrict__ r0_out,         // [nele][NLOC]
    int nele)
{
    __shared__ float stage[8][256];      // 16x16 f32 tile per wave (8 KB/block)

    const int wib = threadIdx.x >> 5;    // wave in block
    // Wave-uniform element index, forced into an SGPR so bases are scalar.
    const int e = __builtin_amdgcn_readfirstlane(blockIdx.x * 8 + wib);
    if (e >= nele) return;               // scalar branch: EXEC stays all-1s

    const int lane = threadIdx.x & 31;
    const int hh   = lane >> 4;          // wave half: 0 or 1
    const int m    = lane & 15;          // A row / B column / D column index
    const int mc   = (m < NLOC) ? m : (NLOC - 1);   // clamped row
    const int mb   = mc * NGI;           // row offset base (elements)

    const float kval = kptr[0];
    const float rdt  = __builtin_amdgcn_rcpf(dtptr[0]);  // 1/dt

    // Per-element scalar bases: every access below is saddr + 32-bit voffset.
    const float* nxe0 = nx     + (size_t)e * (2 * NLOC * NGI);
    const float* nxe1 = nxe0   + NLOC * NGI;
    const float* w    = detwei + (size_t)e * NGI;
    const float* cne  = c_n    + (size_t)e * NLOC;
    const float* cie  = c_i    + (size_t)e * NLOC;
    const float* r0e  = r0_in  + (size_t)e * NLOC;
    float* bout = bdiagA + (size_t)e * (NLOC * NLOC);
    float* dout = diagA  + (size_t)e * NLOC;
    float* rout = r0_out + (size_t)e * NLOC;

    v8f accS = {0.f, 0.f, 0.f, 0.f, 0.f, 0.f, 0.f, 0.f};   // stiffness (pre-k)
    v8f accN = {0.f, 0.f, 0.f, 0.f, 0.f, 0.f, 0.f, 0.f};   // mass

    // k-steps 0..2: all K indices valid (K <= 11), no masking at all.
    #pragma unroll
    for (int s = 0; s < 3; ++s) {
        const int k0 = 4 * s + 2 * hh;       // A: v0=K(k0), v1=K(k0+1); B mirrors
        const float w0 = __builtin_nontemporal_load(&w[k0]);
        const float w1 = __builtin_nontemporal_load(&w[k0 + 1]);

        v2f b0 = { __builtin_nontemporal_load(&nxe0[mb + k0]),
                   __builtin_nontemporal_load(&nxe0[mb + k0 + 1]) };
        v2f a0 = { b0.x * w0, b0.y * w1 };
        accS = __builtin_amdgcn_wmma_f32_16x16x4_f32(
                   false, a0, false, b0, (short)0, accS, false, false);

        v2f b1 = { __builtin_nontemporal_load(&nxe1[mb + k0]),
                   __builtin_nontemporal_load(&nxe1[mb + k0 + 1]) };
        v2f a1 = { b1.x * w0, b1.y * w1 };
        accS = __builtin_amdgcn_wmma_f32_16x16x4_f32(
                   false, a1, false, b1, (short)0, accS, false, false);

        v2f bn = { nmat[mb + k0], nmat[mb + k0 + 1] };   // shared: keep cached
        v2f an = { bn.x * w0, bn.y * w1 };
        accN = __builtin_amdgcn_wmma_f32_16x16x4_f32(
                   false, an, false, bn, (short)0, accN, false, false);
    }

    // k-step 3: K=12..15; only K=12 (half 0, v0) is live.
    {
        const float z   = (hh == 0) ? 1.0f : 0.0f;
        const float w12 = __builtin_nontemporal_load(&w[12]);

        v2f b0 = { __builtin_nontemporal_load(&nxe0[mb + 12]) * z, 0.f };
        v2f a0 = { b0.x * w12, 0.f };
        accS = __builtin_amdgcn_wmma_f32_16x16x4_f32(
                   false, a0, false, b0, (short)0, accS, false, false);

        v2f b1 = { __builtin_nontemporal_load(&nxe1[mb + 12]) * z, 0.f };
        v2f a1 = { b1.x * w12, 0.f };
        accS = __builtin_amdgcn_wmma_f32_16x16x4_f32(
                   false, a1, false, b1, (short)0, accS, false, false);

        v2f bn = { nmat[mb + 12] * z, 0.f };
        v2f an = { bn.x * w12, 0.f };
        accN = __builtin_amdgcn_wmma_f32_16x16x4_f32(
                   false, an, false, bn, (short)0, accN, false, false);
    }

    // Mat-vec operands per D column m (zeroed for padded columns).
    float cn = __builtin_nontemporal_load(&cne[mc]);
    float ci = __builtin_nontemporal_load(&cie[mc]);
    cn = (m < NLOC) ? cn : 0.f;
    ci = (m < NLOC) ? ci : 0.f;

    float* st = &stage[wib][0];

    v8f part;
    #pragma unroll
    for (int r = 0; r < 8; ++r) {
        const float nd = accN[r] * rdt;          // nn / dt
        const float x  = nd + kval * accS[r];    // nxnx final
        part[r] = nd * cn - x * ci;              // per-column contribution
        // Stage D[row][col]: unconditional, immediate LDS offsets.
        st[(r + 8 * hh) * 16 + m] = x;
    }

    // Reduce mat-vec over the 16 columns (xor masks stay within each half).
    #pragma unroll
    for (int r = 0; r < 8; ++r) {
        float p = part[r];
        p += swz_xor<1>(p);
        p += swz_xor<2>(p);
        p += swz_xor<4>(p);
        p += swz_xor<8>(p);
        part[r] = p;
    }

    // bdiagA: copy the 100 valid floats out linearly (coalesced NT stores).
    #pragma unroll
    for (int j = 0; j < 3; ++j) {        // i = 0..95: always valid
        const int i   = lane + 32 * j;
        const int row = i / NLOC;
        const int col = i - row * NLOC;
        __builtin_nontemporal_store(st[row * 16 + col], &bout[i]);
    }
    {                                     // i = 96..99: lanes 0..3 only
        const int i = lane + 96;
        if (lane < 4) {
            const int row = i / NLOC;
            const int col = i - row * NLOC;
            __builtin_nontemporal_store(st[row * 16 + col], &bout[i]);
        }
    }

    // diagA: D[m][m] sits at LDS offset 17*m (conflict-free).
    if (lane < NLOC)
        __builtin_nontemporal_store(st[17 * lane], &dout[lane]);

    // r0: half0 lane r writes row r, half1 lane r writes row r+8.
    const int myrow = m + 8 * hh;
    float mypart = 0.f;
    #pragma unroll
    for (int r = 0; r < 8; ++r)
        if (m == r) mypart = part[r];
    const int rowc = (myrow < NLOC) ? myrow : 0;
    const float r0v = __builtin_nontemporal_load(&r0e[rowc]);
    if (m < 8 && myrow < NLOC)
        __builtin_nontemporal_store(r0v + mypart, &rout[myrow]);
}

extern "C" void kernel_launch(void* const* d_in, const int* in_sizes, int n_in,
                              void* d_out, int out_size, void* d_ws, size_t ws_size,
                              hipStream_t stream) {
    const float* r0_in  = (const float*)d_in[0];
    const float* c_i    = (const float*)d_in[1];
    const float* c_n    = (const float*)d_in[2];
    const float* kptr   = (const float*)d_in[3];
    const float* dtptr  = (const float*)d_in[4];
    const float* nmat   = (const float*)d_in[5];
    const float* nx     = (const float*)d_in[6];
    const float* detwei = (const float*)d_in[7];

    const int nele = in_sizes[0] / NLOC;

    float* bdiagA = (float*)d_out;                                // nele*100
    float* diagA  = bdiagA + (size_t)nele * NLOC * NLOC;          // nele*10
    float* r0_out = diagA  + (size_t)nele * NLOC;                 // nele*10

    const int waves_per_block = 256 / 32;
    const int blocks = (nele + waves_per_block - 1) / waves_per_block;
    fem_assemble<<<blocks, 256, 0, stream>>>(r0_in, c_i, c_n, kptr, dtptr,
                                             nmat, nx, detwei,
                                             bdiagA, diagA, r0_out, nele);
}